// ReconstructionGrid_15238543966484
// MI455X (gfx1250) — compile-verified
//
#include <hip/hip_runtime.h>
#include <math.h>

// Grid resolution from the reference
#define Z_RES 128
#define N_RES 512
#define NSLAB 4
#define SLAB_SHIFT 5          // z0 >> 5 -> 4 slabs of 32 z-planes (+1 boundary)
                              // normal-grid footprint per slab: 33*512*512*12B ~ 99 MB < 192 MB L2

typedef float v4f __attribute__((ext_vector_type(4)));

// 4-byte-aligned aggregates so the compiler can emit wide global_load_b64/bXX
// gathers even though the base offset is only dword-aligned.
struct __attribute__((aligned(4))) PairF  { float v0, v1; };
struct __attribute__((aligned(4))) Pair3F { float v[6]; };

__device__ __forceinline__ float tanh_fast(float x) {
#if __has_builtin(__builtin_amdgcn_tanhf)
    return __builtin_amdgcn_tanhf(x);   // v_tanh_f32 on gfx1250
#else
    return tanhf(x);
#endif
}

// Per-point index/weight computation (shared by all passes).
struct Cell {
    int b00, b01, b10, b11;   // element index of the y-pair base at the 4 (z,x) corners
    float w00, w01, w10, w11; // (z,x) corner weights
    float w0y, w1y;           // y-pair weights (clamp-corrected)
    int slab;                 // z-slab for L2 binning
};

__device__ __forceinline__ Cell decode_c(float cz, float cx, float cy) {
    const float fz0 = floorf(cz), fx0 = floorf(cx), fy0 = floorf(cy);
    const float fz = cz - fz0, fx = cx - fx0, fy = cy - fy0;

    int z0 = min(max((int)fz0, 0), Z_RES - 1);
    int x0 = min(max((int)fx0, 0), N_RES - 1);
    int y0 = min(max((int)fy0, 0), N_RES - 1);
    const int z1 = min(z0 + 1, Z_RES - 1);
    const int x1 = min(x0 + 1, N_RES - 1);

    // y-pair: load elements [yb, yb+1]; when y0 is clamped to the last column
    // the pair re-bases to N-2 and all weight shifts onto element 1.
    const int  yb  = min(y0, N_RES - 2);
    const bool yin = (y0 < N_RES - 1);

    Cell c;
    c.w0y = yin ? (1.0f - fy) : 0.0f;
    c.w1y = yin ? fy : 1.0f;

    const float wz0 = 1.0f - fz, wz1 = fz;
    const float wx0 = 1.0f - fx, wx1 = fx;
    c.w00 = wz0 * wx0;
    c.w01 = wz0 * wx1;
    c.w10 = wz1 * wx0;
    c.w11 = wz1 * wx1;

    c.b00 = (z0 * N_RES + x0) * N_RES + yb;
    c.b01 = (z0 * N_RES + x1) * N_RES + yb;
    c.b10 = (z1 * N_RES + x0) * N_RES + yb;
    c.b11 = (z1 * N_RES + x1) * N_RES + yb;
    c.slab = z0 >> SLAB_SHIFT;
    return c;
}

__device__ __forceinline__ void albedo_point(const Cell& c,
                                             const float* __restrict__ albedo,
                                             float* __restrict__ out_a, int i) {
    const PairF a00 = *(const PairF*)(albedo + c.b00);
    const PairF a01 = *(const PairF*)(albedo + c.b01);
    const PairF a10 = *(const PairF*)(albedo + c.b10);
    const PairF a11 = *(const PairF*)(albedo + c.b11);

    float a = (a00.v0 * c.w0y + a00.v1 * c.w1y) * c.w00
            + (a01.v0 * c.w0y + a01.v1 * c.w1y) * c.w01
            + (a10.v0 * c.w0y + a10.v1 * c.w1y) * c.w10
            + (a11.v0 * c.w0y + a11.v1 * c.w1y) * c.w11;

    a = (a > 0.0f) ? a : expm1f(a);              // ELU(alpha=1)
    __builtin_nontemporal_store(a, out_a + i);   // coalesced stream -> NT
}

// NTSTORE: coalesced output (direct path) uses NT; scattered output (binned
// path) uses RT so L2 can write-coalesce the 64B lines.
template <bool NTSTORE>
__device__ __forceinline__ void normal_point(const Cell& c,
                                             const float* __restrict__ normal,
                                             float* __restrict__ out_n, int pid) {
    const Pair3F p00 = *(const Pair3F*)(normal + c.b00 * 3);
    const Pair3F p01 = *(const Pair3F*)(normal + c.b01 * 3);
    const Pair3F p10 = *(const Pair3F*)(normal + c.b10 * 3);
    const Pair3F p11 = *(const Pair3F*)(normal + c.b11 * 3);

    float t[3];
#pragma unroll
    for (int ch = 0; ch < 3; ++ch) {
        const float v = (p00.v[ch] * c.w0y + p00.v[3 + ch] * c.w1y) * c.w00
                      + (p01.v[ch] * c.w0y + p01.v[3 + ch] * c.w1y) * c.w01
                      + (p10.v[ch] * c.w0y + p10.v[3 + ch] * c.w1y) * c.w10
                      + (p11.v[ch] * c.w0y + p11.v[3 + ch] * c.w1y) * c.w11;
        t[ch] = tanh_fast(v);
    }
    t[0] -= 1.0f;   // base_normal = (-1, 0, 0)

    const float s   = t[0] * t[0] + t[1] * t[1] + t[2] * t[2];
    const float nrm = fmaxf(sqrtf(s), 1e-12f);
    const float inv = 1.0f / nrm;

    if (NTSTORE) {
        __builtin_nontemporal_store(t[0] * inv, out_n + 3 * pid + 0);
        __builtin_nontemporal_store(t[1] * inv, out_n + 3 * pid + 1);
        __builtin_nontemporal_store(t[2] * inv, out_n + 3 * pid + 2);
    } else {
        out_n[3 * pid + 0] = t[0] * inv;
        out_n[3 * pid + 1] = t[1] * inv;
        out_n[3 * pid + 2] = t[2] * inv;
    }
}

// ---------------------------------------------------------------------------
// Pass 1: albedo gather + per-block slab histogram. The 128 MB albedo grid
// fits in the 192 MB L2, and this pass runs alone before anything thrashes it.
__global__ __launch_bounds__(256) void devox_albedo_count_kernel(
    const float* __restrict__ coords, const float* __restrict__ albedo,
    float* __restrict__ out_a, int* __restrict__ counts, int M)
{
    __shared__ int lcnt[NSLAB];
    const int tid = threadIdx.x;
    if (tid < NSLAB) lcnt[tid] = 0;
    __syncthreads();

    const int i = blockIdx.x * 256 + tid;
    if (i < M) {
        const float cz = __builtin_nontemporal_load(coords + 3 * i + 0);
        const float cx = __builtin_nontemporal_load(coords + 3 * i + 1);
        const float cy = __builtin_nontemporal_load(coords + 3 * i + 2);
        const Cell c = decode_c(cz, cx, cy);
        albedo_point(c, albedo, out_a, i);
        atomicAdd(&lcnt[c.slab], 1);
    }
    __syncthreads();
    if (tid < NSLAB) atomicAdd(&counts[tid], lcnt[tid]);
}

__global__ void zero_kernel(int* __restrict__ counts) {
    if (threadIdx.x < NSLAB) counts[threadIdx.x] = 0;
}

__global__ void scan_kernel(const int* __restrict__ counts, int* __restrict__ cursors) {
    if (threadIdx.x == 0) {
        int run = 0;
#pragma unroll
        for (int s = 0; s < NSLAB; ++s) { cursors[s] = run; run += counts[s]; }
    }
}

// Pass 2: scatter {coords, pid} payloads into contiguous slab ranges.
// Hierarchical aggregation: intra-block rank via LDS atomics, then ONE global
// atomicAdd per slab per block (~31K total instead of 2M on 4 hot addresses,
// which would serialize at the L2 atomic units). Per-block payload stores land
// in 4 contiguous runs -> coalesced b128 NT stores.
__global__ __launch_bounds__(256) void scatter_kernel(
    const float* __restrict__ coords, int* __restrict__ cursors,
    v4f* __restrict__ payload, int M)
{
    __shared__ int lcnt[NSLAB];
    __shared__ int lbase[NSLAB];
    const int tid = threadIdx.x;
    if (tid < NSLAB) lcnt[tid] = 0;
    __syncthreads();

    const int i = blockIdx.x * 256 + tid;
    const bool active = (i < M);

    float cz = 0.0f, cx = 0.0f, cy = 0.0f;
    int slab = 0, rank = 0;
    if (active) {
        cz = __builtin_nontemporal_load(coords + 3 * i + 0);
        cx = __builtin_nontemporal_load(coords + 3 * i + 1);
        cy = __builtin_nontemporal_load(coords + 3 * i + 2);
        const int z0 = min(max((int)floorf(cz), 0), Z_RES - 1);
        slab = z0 >> SLAB_SHIFT;
        rank = atomicAdd(&lcnt[slab], 1);            // LDS: intra-block rank
    }
    __syncthreads();
    if (tid < NSLAB) {
        const int n = lcnt[tid];
        lbase[tid] = (n > 0) ? atomicAdd(&cursors[tid], n) : 0;  // 1 global atomic/slab/block
    }
    __syncthreads();
    if (active) {
        v4f p;
        p.x = cz; p.y = cx; p.z = cy; p.w = __int_as_float(i);
        __builtin_nontemporal_store(p, payload + (lbase[slab] + rank));
    }
}

// Pass 3: normal gather in slab order -> per-phase footprint ~99 MB, L2-resident.
__global__ __launch_bounds__(256) void devox_normal_binned_kernel(
    const v4f* __restrict__ payload, const float* __restrict__ normal,
    float* __restrict__ out_n, int M)
{
    const int j = blockIdx.x * 256 + threadIdx.x;
    if (j >= M) return;
    const v4f p = __builtin_nontemporal_load(payload + j);
    const int pid = __float_as_int(p.w);
    const Cell c = decode_c(p.x, p.y, p.z);
    normal_point<false>(c, normal, out_n, pid);      // scattered store -> RT
}

// ---------------------------------------------------------------------------
// Fallback (ws too small): direct two-pass version.
__global__ __launch_bounds__(256) void devox_albedo_kernel(
    const float* __restrict__ coords, const float* __restrict__ albedo,
    float* __restrict__ out_a, int M)
{
    const int i = blockIdx.x * 256 + threadIdx.x;
    if (i >= M) return;
    const float cz = __builtin_nontemporal_load(coords + 3 * i + 0);
    const float cx = __builtin_nontemporal_load(coords + 3 * i + 1);
    const float cy = __builtin_nontemporal_load(coords + 3 * i + 2);
    albedo_point(decode_c(cz, cx, cy), albedo, out_a, i);
}

__global__ __launch_bounds__(256) void devox_normal_kernel(
    const float* __restrict__ coords, const float* __restrict__ normal,
    float* __restrict__ out_n, int M)
{
    const int i = blockIdx.x * 256 + threadIdx.x;
    if (i >= M) return;
    const float cz = __builtin_nontemporal_load(coords + 3 * i + 0);
    const float cx = __builtin_nontemporal_load(coords + 3 * i + 1);
    const float cy = __builtin_nontemporal_load(coords + 3 * i + 2);
    normal_point<true>(decode_c(cz, cx, cy), normal, out_n, i);  // coalesced -> NT
}

// ---------------------------------------------------------------------------
extern "C" void kernel_launch(void* const* d_in, const int* in_sizes, int n_in,
                              void* d_out, int out_size, void* d_ws, size_t ws_size,
                              hipStream_t stream) {
    (void)n_in; (void)out_size;

    const float* coords = (const float*)d_in[0];  // (M, 3)
    const float* albedo = (const float*)d_in[1];  // (Z, N, N)
    const float* normal = (const float*)d_in[2];  // (Z, N, N, 3)

    const int M = in_sizes[0] / 3;
    float* out_a = (float*)d_out;   // (M,)
    float* out_n = out_a + M;       // (M, 3)

    const int blocks = (M + 255) / 256;
    const size_t need = 256 + (size_t)M * sizeof(v4f);

    if (d_ws != nullptr && ws_size >= need) {
        int* counts  = (int*)d_ws;                       // [0..4)
        int* cursors = counts + NSLAB;                   // [4..8)
        v4f* payload = (v4f*)((char*)d_ws + 256);        // 16B-aligned, M entries

        zero_kernel<<<1, 32, 0, stream>>>(counts);
        devox_albedo_count_kernel<<<blocks, 256, 0, stream>>>(coords, albedo, out_a, counts, M);
        scan_kernel<<<1, 32, 0, stream>>>(counts, cursors);
        scatter_kernel<<<blocks, 256, 0, stream>>>(coords, cursors, payload, M);
        devox_normal_binned_kernel<<<blocks, 256, 0, stream>>>(payload, normal, out_n, M);
    } else {
        devox_albedo_kernel<<<blocks, 256, 0, stream>>>(coords, albedo, out_a, M);
        devox_normal_kernel<<<blocks, 256, 0, stream>>>(coords, normal, out_n, M);
    }
}